// HarmonicModel_28965259444499
// MI455X (gfx1250) — compile-verified
//
#include <hip/hip_runtime.h>
#include <math.h>

#define SRATE_F     16000.0f
#define FRAME_STEP  64
#define CHANNELS    4
#define FRAMES      1000
#define HARMONICS   64
#define TWO_PI_F    6.283185307179586f
#define INV_TWO_PI  0.15915494309189535f

typedef __attribute__((ext_vector_type(2))) float v2f;
typedef __attribute__((ext_vector_type(8))) float v8f;

// Bounded-range sine: phase is guaranteed in [0, ~200] rad, so reduce the
// period explicitly (3 VALU ops) and use the hardware transcendental
// (v_sin_f32 via __sinf) instead of libm's Payne-Hanek slow path.
__device__ __forceinline__ float fast_sin_bounded(float ph) {
    float r = ph * INV_TWO_PI;
    r -= floorf(r);                 // [0,1)
    return __sinf(r * TWO_PI_F);    // v_sin_f32
}

// ---------------------------------------------------------------------------
// Kernel 1: frame-level phase integration (exclusive cumsum of per-frame
// phase increments, wrapped to [0,2pi) — matches jnp unwrapped-f32-cumsum
// then mod semantics). 256 independent series of length 1000; one thread each.
// ---------------------------------------------------------------------------
__global__ void phase_scan_kernel(const float* __restrict__ h_freq,
                                  const float* __restrict__ shifts,
                                  float* __restrict__ phase) {
    int t = blockIdx.x * blockDim.x + threadIdx.x;   // 0..255
    if (t >= CHANNELS * HARMONICS) return;
    int c = t >> 6;
    int h = t & 63;
    const float frame_rate = SRATE_F / (float)FRAME_STEP;              // 250
    const float k_dph = TWO_PI_F * ((float)FRAME_STEP / SRATE_F);      // 2pi*0.004
    float acc = 0.0f;                                                  // unwrapped f32 cumsum, like reference
    int base = c * FRAMES * HARMONICS + h;
    for (int k = 0; k < FRAMES; ++k) {
        int idx = base + k * HARMONICS;
        float f = h_freq[idx] + frame_rate * shifts[idx * 3 + 0];
        // exclusive: write phase of frame k BEFORE adding frame k's increment
        phase[idx] = acc - floorf(acc * INV_TWO_PI) * TWO_PI_F;
        acc += k_dph * f;
    }
}

// ---------------------------------------------------------------------------
// Kernel 2: additive synthesis. One wave per 16-sample tile of a frame.
// Per WMMA K-chunk kk (16 chunks cover 64 harmonics):
//   A[M=sample(16) x K=4]  = m_seg * sin(ph)   (f32 products, ISA A-layout)
//   B[4 x 16]              = ones
//   D = A@B + C            -> every D column accumulates the harmonic sum
// Closed-form within-frame phase (dph is linear in s, so its exclusive
// cumsum is  a*s + b*s(s-1)/2 ) — no inner scan.
// ---------------------------------------------------------------------------
__global__ void __launch_bounds__(256)
harmonic_synth_kernel(const float* __restrict__ h_freq,
                      const float* __restrict__ h_mag,
                      const float* __restrict__ shifts,
                      const float* __restrict__ phase,
                      float* __restrict__ out) {
    const int wave = threadIdx.x >> 5;
    const int lane = threadIdx.x & 31;
    const int tile = blockIdx.x * 8 + wave;          // 0..15999, 4 tiles/frame
    const int c    = tile / (FRAMES * 4);
    const int r    = tile - c * (FRAMES * 4);
    const int k    = r >> 2;
    const int tseg = r & 3;
    const int k1   = (k < FRAMES - 1) ? (k + 1) : k; // hold last frame

    const int   row  = lane & 15;                    // M = sample row
    const int   hoff = (lane >> 4) << 1;             // K sub-offset 0 / 2
    const int   s    = tseg * 16 + row;              // sample within frame
    const float sf   = (float)s;
    const float w    = sf * (1.0f / (float)FRAME_STEP);
    const float tri  = 0.5f * sf * (sf - 1.0f);      // s(s-1)/2

    const int base0 = (c * FRAMES + k)  * HARMONICS;
    const int base1 = (c * FRAMES + k1) * HARMONICS;
    const float frame_rate = SRATE_F / (float)FRAME_STEP;
    const float c2pi_sr    = TWO_PI_F / SRATE_F;

    v8f acc = {};
    v2f bones; bones[0] = 1.0f; bones[1] = 1.0f;

    for (int kk = 0; kk < 16; ++kk) {
        v2f afrag;
#pragma unroll
        for (int j = 0; j < 2; ++j) {
            const int h  = (kk << 2) + hoff + j;
            const int i0 = base0 + h;
            const int i1 = base1 + h;
            float f0 = h_freq[i0] + frame_rate * shifts[i0 * 3 + 0];
            float f1 = h_freq[i1] + frame_rate * shifts[i1 * 3 + 0];
            float m0 = h_mag[i0]  + shifts[i0 * 3 + 1];
            float m1 = h_mag[i1]  + shifts[i1 * 3 + 1];
            // ph(s) = ph0 + (2pi/SR) * ( f0*s + (f1-f0) * s(s-1)/(2*64) )
            float ph = phase[i0] +
                       c2pi_sr * (f0 * sf + (f1 - f0) * (tri * (1.0f / (float)FRAME_STEP)));
            float ms = m0 * (1.0f - w) + m1 * w;
            afrag[j] = ms * fast_sin_bounded(ph);
        }
        // D = A(16x4 f32) @ ones(4x16) + C  — accumulates harmonic sum
        acc = __builtin_amdgcn_wmma_f32_16x16x4_f32(
            false, afrag, false, bones, (short)0, acc, false, false);
    }

    // D layout: VGPR v, lanes 0-15 -> M=v ; lanes 16-31 -> M=8+v.
    // All columns are equal; lanes 0 and 16 write the 16 samples.
    if ((lane & 15) == 0) {
        const int mbase = (lane >> 4) << 3;          // 0 or 8
        float* o = out + (c * (FRAMES * FRAME_STEP) + k * FRAME_STEP + tseg * 16 + mbase);
#pragma unroll
        for (int v = 0; v < 8; ++v) o[v] = acc[v];
    }
}

extern "C" void kernel_launch(void* const* d_in, const int* in_sizes, int n_in,
                              void* d_out, int out_size, void* d_ws, size_t ws_size,
                              hipStream_t stream) {
    (void)in_sizes; (void)n_in; (void)out_size; (void)ws_size;
    const float* h_freq = (const float*)d_in[0];
    const float* h_mag  = (const float*)d_in[1];
    // d_in[2] = h_phase: dead on the generate_phase=True path
    const float* shifts = (const float*)d_in[3];
    float* phase = (float*)d_ws;                 // 4*1000*64 floats = 1 MB scratch
    float* out   = (float*)d_out;

    phase_scan_kernel<<<1, 256, 0, stream>>>(h_freq, shifts, phase);
    harmonic_synth_kernel<<<2000, 256, 0, stream>>>(h_freq, h_mag, shifts, phase, out);
}